// SpectralConvProd2d_30322469110509
// MI455X (gfx1250) — compile-verified
//
#include <hip/hip_runtime.h>
#include <math.h>

// Problem constants (from reference)
#define B_        8
#define CIN       64
#define COUT      64
#define COUT_LIN  60
#define H_        256
#define W_        256

#define XSTRIDE   260   // padded x-slab row stride (floats): bank-conflict-free,
                        // keeps 16B alignment (260*4 = 1040 = 65*16)
#define TSTRIDE   34    // interleaved (tr,ti) row stride in inverse kernel

typedef float v2f __attribute__((ext_vector_type(2)));
typedef float v8f __attribute__((ext_vector_type(8)));
typedef int   v4i __attribute__((vector_size(16)));   // matches builtin param

#define AS1 __attribute__((address_space(1)))
#define AS3 __attribute__((address_space(3)))

#define TWO_PI_256 0.024543692606170f   // 2*pi/256

// Optional gfx1250 async global->LDS path (probe-confirmed to exist here).
#if defined(__has_builtin)
#if __has_builtin(__builtin_amdgcn_global_load_async_to_lds_b128) && \
    __has_builtin(__builtin_amdgcn_s_wait_asynccnt)
#define HAVE_ASYNC_LDS 1
#endif
#endif

// D = A(16x4) * B(4x16) + C, fp32, wave32 WMMA
static __device__ __forceinline__ v8f wmma_f32(v2f a, v2f b, v8f acc) {
  return __builtin_amdgcn_wmma_f32_16x16x4_f32(false, a, false, b, (short)0, acc,
                                               false, false);
}

// All DFT angles are multiples of 2*pi/256. 256-entry LDS sine LUT:
//   sin(p) = tab[p], cos(p) = tab[p+64], -sin(p) = tab[p+128], -cos = tab[p+192]
#define LUT(tab, p) (tab)[(p) & 255]

// =====================================================================
// Kernel 1: forward partial DFT.  One workgroup per (b,ci), 256 thr.
//   T[h][kx]  = sum_w x[h][w] e^{-2pi i kx w/256}       (real GEMM)
//   X[ky][kx] = sum_h e^{-2pi i ky h/256} T[h][kx]      (complex GEMM)
// ky in {0..15} (low block) and {240..255} (high block).
// =====================================================================
__global__ void fwd_modes(const float* __restrict__ x,
                          float* __restrict__ Xr, float* __restrict__ Xi) {
  extern __shared__ float lds[];
  float* Tr     = lds;                    // 256*16
  float* Ti     = lds + 4096;             // 256*16
  float* xs     = lds + 8192;             // 64*XSTRIDE padded slab
  float* sintab = lds + 8192 + 64 * XSTRIDE;   // 256
  const int tid  = threadIdx.x;
  const int lane = tid & 31;
  const int wave = __builtin_amdgcn_readfirstlane(tid >> 5);  // scalar!
  const int koff = (lane >> 4) << 1;      // K-pair offset per ISA A/B layout
  const int m    = lane & 15;
  const int n    = lane & 15;
  const int bc   = blockIdx.x;            // b*CIN + ci
  const float* xp = x + (size_t)bc * (H_ * W_);

  sintab[tid & 255] = __sinf((float)(tid & 255) * TWO_PI_256);

  // ---- stage 1a: 4 slabs of 64 rows; 8 waves = 4 h-tiles x {re,im}
  const int tile_in_slab = wave >> 1;     // scalar
  const int part         = wave & 1;      // scalar: 0=re, 1=im
  const int poff         = part ? 128 : 64;  // -sin == sin(p+128), cos == sin(p+64)
  for (int s = 0; s < 4; ++s) {
    // copy slab, padded rows: chunk ch -> row=ch>>6, j=ch&63 (16B units)
#if HAVE_ASYNC_LDS
    for (int i = 0; i < 16; ++i) {
      const int ch  = i * 256 + tid;
      const int row = ch >> 6, j = ch & 63;
      __builtin_amdgcn_global_load_async_to_lds_b128(
          (AS1 v4i*)(xp + s * 64 * W_ + row * W_ + j * 4),
          (AS3 v4i*)(xs + row * XSTRIDE + j * 4), 0, 0);
    }
    __builtin_amdgcn_s_wait_asynccnt(0);
#else
    for (int ch = tid; ch < 4096; ch += 256) {
      const int row = ch >> 6, j = ch & 63;
      *(float4*)(xs + row * XSTRIDE + j * 4) =
          *(const float4*)(xp + s * 64 * W_ + row * W_ + j * 4);
    }
#endif
    if (s + 1 < 4) __builtin_prefetch(xp + (s + 1) * 64 * W_ + tid * 64, 0, 0);
    __syncthreads();

    const int hloc = tile_in_slab * 16 + m;
    v8f acc = {};
    for (int k0 = 0; k0 < W_; k0 += 4) {
      v2f a;                               // A = x tile [16h x 4w]
      a.x = xs[hloc * XSTRIDE + k0 + koff];
      a.y = xs[hloc * XSTRIDE + k0 + koff + 1];
      const int p0 = (n * (k0 + koff)) & 255;   // phase kx*w
      const int p1 = p0 + n;
      v2f b; b.x = LUT(sintab, p0 + poff); b.y = LUT(sintab, p1 + poff);
      acc = wmma_f32(a, b, acc);
    }
    float* T = part ? Ti : Tr;
    const int hb = s * 64 + tile_in_slab * 16 + ((lane >> 4) << 3);
#pragma unroll
    for (int r = 0; r < 8; ++r) T[(hb + r) * 16 + n] = acc[r];
    __syncthreads();
  }

  // ---- stage 1b: X = E * T (complex), K = 256 over h. 4 wave-jobs.
  //   Xr = sum cos*Tr + sin*Ti ;  Xi = sum cos*Ti + (-sin)*Tr
  if (wave < 4) {                          // scalar branch
    const int mt = wave >> 1;              // 0=low block, 1=high block
    const int pt = wave & 1;               // 0 -> Xr, 1 -> Xi
    const int ky = (mt ? 240 : 0) + m;
    const float* Bc = pt ? Ti : Tr;        // partner of cos (offset +64)
    const float* Bo = pt ? Tr : Ti;        // partner of sin-term
    const int ooff  = pt ? 128 : 0;        // pt0: +sin (0), pt1: -sin (128)
    v8f acc = {};
    for (int k0 = 0; k0 < H_; k0 += 4) {
      const int h0 = k0 + koff;
      const int p0 = (ky * h0) & 255;      // phase ky*h
      const int p1 = p0 + ky;
      v2f bc; bc.x = Bc[h0 * 16 + n]; bc.y = Bc[(h0 + 1) * 16 + n];
      v2f bo; bo.x = Bo[h0 * 16 + n]; bo.y = Bo[(h0 + 1) * 16 + n];
      v2f ac; ac.x = LUT(sintab, p0 + 64);   ac.y = LUT(sintab, p1 + 64);
      v2f ao; ao.x = LUT(sintab, p0 + ooff); ao.y = LUT(sintab, p1 + ooff);
      acc = wmma_f32(ac, bc, acc);
      acc = wmma_f32(ao, bo, acc);
    }
    float* X = pt ? Xi : Xr;
    const int Mrow = mt * 16 + ((lane >> 4) << 3);
#pragma unroll
    for (int r = 0; r < 8; ++r)
      X[(size_t)bc * 512 + (Mrow + r) * 16 + n] = acc[r];
  }
}

// =====================================================================
// Kernel 2: per-frequency channel mixing + product channels (~0.13 GFLOP,
// VALU). One workgroup per frequency point (2 blocks x 16 x 16 = 512).
// =====================================================================
__global__ void mix_prod(const float* __restrict__ Xr, const float* __restrict__ Xi,
                         const float* __restrict__ w1r, const float* __restrict__ w1i,
                         const float* __restrict__ w2r, const float* __restrict__ w2i,
                         float* __restrict__ Yr, float* __restrict__ Yi) {
  __shared__ float Xsr[B_ * CIN], Xsi[B_ * CIN];
  __shared__ float Ysr[B_ * COUT], Ysi[B_ * COUT];
  const int p   = blockIdx.x;        // blk*256 + ky*16 + kx
  const int blk = p >> 8;
  const int off = p & 255;
  const int t = threadIdx.x;         // 512 threads: (b, channel)
  const int b = t >> 6, c = t & 63;

  Xsr[t] = Xr[(size_t)(b * CIN + c) * 512 + blk * 256 + off];
  Xsi[t] = Xi[(size_t)(b * CIN + c) * 512 + blk * 256 + off];
  __syncthreads();

  const float* wr = blk ? w2r : w1r;
  const float* wi = blk ? w2i : w1i;
  if (c < COUT_LIN) {
    float ar = 0.f, ai = 0.f;
    for (int ci = 0; ci < CIN; ++ci) {
      const float xr = Xsr[b * CIN + ci], xi = Xsi[b * CIN + ci];
      const float vr = wr[(size_t)(ci * COUT_LIN + c) * 256 + off];
      const float vi = wi[(size_t)(ci * COUT_LIN + c) * 256 + off];
      ar += xr * vr - xi * vi;
      ai += xr * vi + xi * vr;
    }
    Ysr[t] = ar; Ysi[t] = ai;
  }
  __syncthreads();
  if (c >= COUT_LIN) {                     // product channels 60..63
    const int np = c - COUT_LIN;
    const float ar = Ysr[b * COUT + 2 * np],     ai = Ysi[b * COUT + 2 * np];
    const float br = Ysr[b * COUT + 2 * np + 1], bi = Ysi[b * COUT + 2 * np + 1];
    Ysr[t] = ar * br - ai * bi;
    Ysi[t] = ar * bi + ai * br;
  }
  Yr[(size_t)(b * COUT + c) * 512 + blk * 256 + off] = Ysr[t];
  Yi[(size_t)(b * COUT + c) * 512 + blk * 256 + off] = Ysi[t];
}

// =====================================================================
// Kernel 3: inverse. One workgroup per (b,co), 256 thr.
//   tmp[h][kx] = s_kx * sum_{ky in S} Y[ky][kx] e^{+2pi i ky h/256}
//     (s_0 = 1/HW, s_kx = 2/HW; imag of kx=0 zeroed -> scales folded here)
//   out[h][w]  = sum_kx (Re tmp * cos + Im tmp * (-sin))
// =====================================================================
__global__ void inv_modes(const float* __restrict__ Yr, const float* __restrict__ Yi,
                          float* __restrict__ out) {
  extern __shared__ float lds[];
  float* tp     = lds;                    // 256 rows x TSTRIDE, (tr,ti) pairs
  float* ylr    = lds + 256 * TSTRIDE;    // 32*16
  float* yli    = ylr + 512;              // 32*16
  float* sintab = yli + 512;              // 256
  const int tid  = threadIdx.x;
  const int lane = tid & 31;
  const int wave = __builtin_amdgcn_readfirstlane(tid >> 5);  // scalar!
  const int koff = (lane >> 4) << 1;
  const int m = lane & 15, n = lane & 15;
  const int bc = blockIdx.x;     // b*COUT + co

  sintab[tid] = __sinf((float)tid * TWO_PI_256);
  if (tid < 128) {
    ((float4*)ylr)[tid] = ((const float4*)(Yr + (size_t)bc * 512))[tid];
    ((float4*)yli)[tid] = ((const float4*)(Yi + (size_t)bc * 512))[tid];
  }
  __syncthreads();

  // stage A: 32 jobs = 16 h-tiles x {re,im}; K=32 over ky-index
  //   tr = cos*Yr + (-sin)*Yi ;  ti = cos*Yi + sin*Yr
  const float inv = 1.0f / (float)(H_ * W_);
  for (int job = wave; job < 32; job += 8) {
    const int mt = job >> 1, pt = job & 1;          // scalar
    const float* Bc = pt ? yli : ylr;
    const float* Bo = pt ? ylr : yli;
    const int ooff  = pt ? 0 : 128;                 // pt0: -sin, pt1: +sin
    const int h = mt * 16 + m;
    v8f acc = {};
#pragma unroll
    for (int k0 = 0; k0 < 32; k0 += 4) {
      const int kk0 = k0 + koff;
      const int ky0 = (kk0 < 16) ? kk0 : (kk0 + 224);  // high block 240..255
      const int p0 = (ky0 * h) & 255;
      const int p1 = p0 + h;                           // (ky0+1)*h
      v2f bc; bc.x = Bc[kk0 * 16 + n]; bc.y = Bc[(kk0 + 1) * 16 + n];
      v2f bo; bo.x = Bo[kk0 * 16 + n]; bo.y = Bo[(kk0 + 1) * 16 + n];
      v2f ac; ac.x = LUT(sintab, p0 + 64);   ac.y = LUT(sintab, p1 + 64);
      v2f ao; ao.x = LUT(sintab, p0 + ooff); ao.y = LUT(sintab, p1 + ooff);
      acc = wmma_f32(ac, bc, acc);
      acc = wmma_f32(ao, bo, acc);
    }
    // fold irfft scale s_kx (and kx=0 imag drop) into the store; n == kx
    const float scl = pt ? ((n == 0) ? 0.f : 2.f * inv)
                         : ((n == 0) ? inv : 2.f * inv);
    const int hb = mt * 16 + ((lane >> 4) << 3);
#pragma unroll
    for (int r = 0; r < 8; ++r)
      tp[(hb + r) * TSTRIDE + n * 2 + pt] = acc[r] * scl;
  }
  __syncthreads();

  // stage B: reconstruction GEMM [256h x 32] * [32 x 256w]
  //   A pair (tr,ti) is one b64 load; B = (cos, -sin) pure LUT reads.
  float* op = out + (size_t)bc * (H_ * W_);
  for (int tile = wave; tile < 256; tile += 8) {
    const int ht = tile >> 4, wt = tile & 15;
    const int w  = wt * 16 + n;
    v8f acc = {};
#pragma unroll
    for (int k0 = 0; k0 < 32; k0 += 4) {
      const int kx = (k0 + koff) >> 1;     // q0 even: x=re, y=im, same kx
      const int h  = ht * 16 + m;
      v2f a = *(const v2f*)(tp + h * TSTRIDE + kx * 2);
      const int p = (kx * w) & 255;
      v2f b; b.x = LUT(sintab, p + 64); b.y = LUT(sintab, p + 128);
      acc = wmma_f32(a, b, acc);
    }
    const int hb = ht * 16 + ((lane >> 4) << 3);
#pragma unroll
    for (int r = 0; r < 8; ++r)
      op[(size_t)(hb + r) * W_ + w] = acc[r];
  }
}

extern "C" void kernel_launch(void* const* d_in, const int* in_sizes, int n_in,
                              void* d_out, int out_size, void* d_ws, size_t ws_size,
                              hipStream_t stream) {
  (void)in_sizes; (void)n_in; (void)out_size; (void)ws_size;
  const float* x   = (const float*)d_in[0];
  const float* w1r = (const float*)d_in[1];
  const float* w1i = (const float*)d_in[2];
  const float* w2r = (const float*)d_in[3];
  const float* w2i = (const float*)d_in[4];
  float* out = (float*)d_out;

  // workspace: forward modes X and mixed modes Y, each [.,.,32,16] re/im
  float* Xr = (float*)d_ws;
  float* Xi = Xr + (size_t)B_ * CIN * 512;
  float* Yr = Xi + (size_t)B_ * CIN * 512;
  float* Yi = Yr + (size_t)B_ * COUT * 512;

  const int lds1 = (8192 + 64 * XSTRIDE + 256) * 4;            // fwd_modes
  const int lds3 = (256 * TSTRIDE + 512 + 512 + 256) * 4;      // inv_modes
  fwd_modes<<<dim3(B_ * CIN), dim3(256), lds1, stream>>>(x, Xr, Xi);
  mix_prod <<<dim3(512),      dim3(512), 0,    stream>>>(Xr, Xi, w1r, w1i,
                                                         w2r, w2i, Yr, Yi);
  inv_modes<<<dim3(B_ * COUT), dim3(256), lds3, stream>>>(Yr, Yi, out);
}